// LIFSpike_20899310862881
// MI455X (gfx1250) — compile-verified
//
#include <hip/hip_runtime.h>

// LIF spike forward on MI455X (gfx1250).
// x: [T=8, B*C*H*W = 8,388,608] fp32, out: same shape (spikes in {0,1}).
// Pure HBM-streaming kernel: 512 MiB traffic, ~23 us floor @ 23.3 TB/s.
// One thread owns 4 consecutive spatial elements (float4 column) across all
// 8 timesteps: 8x global_load_b128 (NT) issued up front, recurrence in
// registers, 8x global_store_b128 (NT).

typedef __attribute__((ext_vector_type(4))) float v4f;

#define LIF_T     8
#define LIF_TAU   0.25f
#define LIF_TH    1.0f

__global__ __launch_bounds__(256) void lif_spike_fwd(const float* __restrict__ xg,
                                                     float* __restrict__ og,
                                                     long n4 /* float4s per timestep */) {
  const long i = (long)blockIdx.x * blockDim.x + threadIdx.x;
  if (i >= n4) return;

  const v4f* __restrict__ x = (const v4f*)xg;
  v4f* __restrict__ o = (v4f*)og;

  // Issue all T loads up front: 8 outstanding b128 loads per thread,
  // non-temporal (data is single-use; keep the 192MB L2 clean).
  v4f xt[LIF_T];
#pragma unroll
  for (int t = 0; t < LIF_T; ++t) {
    xt[t] = __builtin_nontemporal_load(&x[(long)t * n4 + i]);
  }

  // Sequential LIF recurrence per component (inherently serial in T,
  // nonlinear reset => no scan/matmul reformulation possible).
  v4f m = (v4f)0.0f;
#pragma unroll
  for (int t = 0; t < LIF_T; ++t) {
    v4f s;
#pragma unroll
    for (int k = 0; k < 4; ++k) {
      float mk = LIF_TAU * m[k] + xt[t][k];   // v_fmac_f32
      const bool sp = (mk > LIF_TH);          // v_cmp_gt_f32 (heaviside(mem - th))
      s[k] = sp ? 1.0f : 0.0f;                // v_cndmask_b32
      m[k] = sp ? 0.0f : mk;                  // v_cndmask_b32 (reset)
    }
    // Output is write-once streaming -> non-temporal b128 store.
    __builtin_nontemporal_store(s, &o[(long)t * n4 + i]);
  }
}

extern "C" void kernel_launch(void* const* d_in, const int* in_sizes, int n_in,
                              void* d_out, int out_size, void* d_ws, size_t ws_size,
                              hipStream_t stream) {
  const float* x = (const float*)d_in[0];
  float* out = (float*)d_out;

  // in_sizes[0] = T * B * C * H * W = 8 * 8,388,608 = 67,108,864 floats.
  const long total = (long)in_sizes[0];
  const long per_t = total / LIF_T;   // 8,388,608 (divisible by 4)
  const long n4    = per_t / 4;       // 2,097,152 float4 columns

  const int block = 256;              // 8 wave32s per block
  const long grid = (n4 + block - 1) / block;  // 8192 blocks

  lif_spike_fwd<<<(dim3)(unsigned)grid, (dim3)block, 0, stream>>>(x, out, n4);
}